// SubgraphGINModel_84361747628046
// MI455X (gfx1250) — compile-verified
//
#include <hip/hip_runtime.h>
#include <hip/hip_bf16.h>

#define N_NODES   100000
#define N_EDGES   3200000
#define NUM_GRAPHS 128
#define HIDDEN    32

typedef __attribute__((ext_vector_type(16))) _Float16 v16h;
typedef __attribute__((ext_vector_type(8)))  float    v8f;
typedef __attribute__((ext_vector_type(2)))  float    v2f;

// Guaranteed hardware float-atomic-add at device scope (L2 RMW, no return ->
// STOREcnt). Avoids any chance of a CAS-loop expansion on the 200M+ atomic
// hot path.
__device__ __forceinline__ void atomic_add_f32(float* p, float v) {
    asm volatile("global_atomic_add_f32 %0, %1, off scope:SCOPE_DEV"
                 :
                 : "v"(p), "v"(v)
                 : "memory");
}

// ---- Layer 1 scatter: agg1[dst] += x[src] (scalar per edge) -----------------
__global__ void k_scatter1(const float* __restrict__ x,
                           const long long* __restrict__ ei,
                           float* __restrict__ agg1) {
    long long t = (long long)blockIdx.x * blockDim.x + threadIdx.x;
    if (t >= N_EDGES) return;
    long long s = ei[t];
    long long d = ei[(long long)N_EDGES + t];
    atomic_add_f32(&agg1[d], x[s]);
}

// ---- Layer 1 MLP: h1[i,c] = relu((x[i]+agg1[i])*W1[c] + b1[c]) --------------
__global__ void k_layer1(const float* __restrict__ x,
                         const float* __restrict__ agg1,
                         const float* __restrict__ W1,
                         const float* __restrict__ b1,
                         float* __restrict__ h1) {
    int t = blockIdx.x * blockDim.x + threadIdx.x;   // N_NODES*32 threads
    if (t >= N_NODES * HIDDEN) return;
    int i = t >> 5, c = t & 31;
    float s = x[i] + agg1[i];
    float h = s * W1[c] + b1[c];
    h1[t] = h > 0.f ? h : 0.f;                       // relu (outer relu idempotent)
}

// ---- Layer 2 scatter: agg2[dst,:] += h1[src,:], one wave per edge -----------
__global__ void k_scatter2(const float* __restrict__ h1,
                           const long long* __restrict__ ei,
                           float* __restrict__ agg2) {
    int lane = threadIdx.x & 31;
    long long e = ((long long)blockIdx.x * blockDim.x + threadIdx.x) >> 5;
    if (e >= N_EDGES) return;
    if (lane == 0) {
        __builtin_prefetch(ei + e + 2048, 0, 1);                      // src stream
        __builtin_prefetch(ei + (long long)N_EDGES + e + 2048, 0, 1); // dst stream
    }
    long long s = ei[e];
    long long d = ei[(long long)N_EDGES + e];
    atomic_add_f32(&agg2[d * 32 + lane], h1[s * 32 + lane]);
}

// ---- Layer 2 GEMM (WMMA) + ReLU + global_add_pool, one wave per 16 nodes ----
__global__ void k_layer2_pool(const float* __restrict__ h1,
                              const float* __restrict__ agg2,
                              const float* __restrict__ W2,   // [32][32] row-major (k,n)
                              const float* __restrict__ b2,
                              const long long* __restrict__ batch,
                              float* __restrict__ pool) {     // [128][32]
    int lane = threadIdx.x & 31;
    int wave = (blockIdx.x * blockDim.x + threadIdx.x) >> 5;
    if (wave >= N_NODES / 16) return;                // wave-uniform: EXEC all-1s for WMMA
    int m  = lane & 15;                              // row within tile (A) / col (B)
    int kh = lane >> 4;                              // lane half selects K slice
    int row = wave * 16 + m;

    v8f c0 = {};   // output cols 0..15
    v8f c1 = {};   // output cols 16..31

    // Full f32 WMMA: 8 steps of K=4. A 16x4 layout: lane row m=lane&15,
    // VGPR j holds K = kh*2 + j; B symmetric with col n=lane&15.
    #pragma unroll
    for (int kk = 0; kk < 8; ++kk) {
        int k0 = kk * 4 + kh * 2;
        v2f a, bA, bB;
        a[0]  = h1[row * 32 + k0]     + agg2[row * 32 + k0];
        a[1]  = h1[row * 32 + k0 + 1] + agg2[row * 32 + k0 + 1];
        bA[0] = W2[k0 * 32 + m];        bA[1] = W2[(k0 + 1) * 32 + m];
        bB[0] = W2[k0 * 32 + 16 + m];   bB[1] = W2[(k0 + 1) * 32 + 16 + m];
        c0 = __builtin_amdgcn_wmma_f32_16x16x4_f32(false, a, false, bA, (short)0, c0, false, false);
        c1 = __builtin_amdgcn_wmma_f32_16x16x4_f32(false, a, false, bB, (short)0, c1, false, false);
    }

    // Epilogue: +bias, relu, pool by graph id.  C/D layout: element (v + 8*kh, lane&15).
    int   n0 = m, n1 = 16 + m;
    float bb0 = b2[n0], bb1 = b2[n1];
    #pragma unroll
    for (int v = 0; v < 8; ++v) {
        int mm   = v + 8 * kh;
        int grow = wave * 16 + mm;
        long long g = batch[grow];
        float h2a = c0[v] + bb0; h2a = h2a > 0.f ? h2a : 0.f;
        float h2b = c1[v] + bb1; h2b = h2b > 0.f ? h2b : 0.f;
        atomic_add_f32(&pool[g * 32 + n0], h2a);
        atomic_add_f32(&pool[g * 32 + n1], h2b);
    }
}

// ---- Final: out = |pool1 - pool2| -------------------------------------------
__global__ void k_absdiff(const float* __restrict__ p1,
                          const float* __restrict__ p2,
                          float* __restrict__ out) {
    int t = blockIdx.x * blockDim.x + threadIdx.x;
    if (t < NUM_GRAPHS * HIDDEN) out[t] = fabsf(p1[t] - p2[t]);
}

// Workspace layout per graph (floats):
//   agg1 @ 0            (N)
//   agg2 @ N            (32N)
//   pool @ 33N          (4096)
//   h1   @ 33N + 4096   (32N)
// region size = 65N + 4096 floats; [agg1|agg2|pool] contiguous -> one memset.
static void encode_graph(const float* x, const long long* ei, const long long* batch,
                         const float* W1, const float* b1,
                         const float* W2, const float* b2,
                         float* region, hipStream_t stream) {
    const long long N = N_NODES;
    float* agg1 = region;
    float* agg2 = region + N;
    float* pool = region + 33 * N;
    float* h1   = region + 33 * N + 4096;

    hipMemsetAsync(region, 0, (size_t)(33 * N + 4096) * sizeof(float), stream);

    k_scatter1<<<(N_EDGES + 255) / 256, 256, 0, stream>>>(x, ei, agg1);
    k_layer1<<<(N_NODES * HIDDEN + 255) / 256, 256, 0, stream>>>(x, agg1, W1, b1, h1);
    k_scatter2<<<((long long)N_EDGES * 32 + 255) / 256, 256, 0, stream>>>(h1, ei, agg2);
    k_layer2_pool<<<((N_NODES / 16) * 32 + 255) / 256, 256, 0, stream>>>(h1, agg2, W2, b2, batch, pool);
}

extern "C" void kernel_launch(void* const* d_in, const int* in_sizes, int n_in,
                              void* d_out, int out_size, void* d_ws, size_t ws_size,
                              hipStream_t stream) {
    const float*     x1  = (const float*)d_in[0];
    const long long* ei1 = (const long long*)d_in[1];
    const long long* bt1 = (const long long*)d_in[2];
    const float*     x2  = (const float*)d_in[3];
    const long long* ei2 = (const long long*)d_in[4];
    const long long* bt2 = (const long long*)d_in[5];
    const float*     W1  = (const float*)d_in[6];
    const float*     b1  = (const float*)d_in[7];
    const float*     W2  = (const float*)d_in[8];
    const float*     b2  = (const float*)d_in[9];
    float* out = (float*)d_out;

    const size_t regionF = 65ull * N_NODES + 4096ull;
    float* r1 = (float*)d_ws;
    float* r2 = r1 + regionF;

    encode_graph(x1, ei1, bt1, W1, b1, W2, b2, r1, stream);
    encode_graph(x2, ei2, bt2, W1, b1, W2, b2, r2, stream);

    const float* pool1 = r1 + 33ull * N_NODES;
    const float* pool2 = r2 + 33ull * N_NODES;
    k_absdiff<<<(NUM_GRAPHS * HIDDEN + 255) / 256, 256, 0, stream>>>(pool1, pool2, out);
}